// PatchNorm_72095321030973
// MI455X (gfx1250) — compile-verified
//
#include <hip/hip_runtime.h>
#include <stdint.h>

// ---------------- problem dims (from reference setup_inputs) ----------------
static constexpr int   B_  = 32;
static constexpr int   S_  = 1024;
static constexpr int   N_  = B_ * S_;      // 32768 tokens
static constexpr int   H_  = 32;
static constexpr int   W_  = 32;
static constexpr int   HW_ = H_ * W_;      // 1024 grid cells
static constexpr int   C_  = 3;
static constexpr int   P2_ = 256;
static constexpr int   D_  = C_ * P2_;     // 768
static constexpr int   TPB_TOK = 8;        // tokens per block in pipelined kernels
#define EPSF 1e-6f

// ---------------- workspace layout (float units) ----------------
static constexpr size_t WS_COUNTS  = 0;                               // HW
static constexpr size_t WS_NNEW    = WS_COUNTS  + HW_;                // HW
static constexpr size_t WS_MEANACC = WS_NNEW    + HW_;                // HW*P2
static constexpr size_t WS_MEANNEW = WS_MEANACC + (size_t)HW_ * P2_;  // HW*P2
static constexpr size_t WS_M2ACC   = WS_MEANNEW + (size_t)HW_ * P2_;  // HW*P2
static constexpr size_t WS_INVSTD  = WS_M2ACC   + (size_t)HW_ * P2_;  // HW*P2
static constexpr size_t WS_TOTAL   = WS_INVSTD  + (size_t)HW_ * P2_;  // ~4.2 MB

// ---------------- CDNA5 async global->LDS (ASYNCcnt-tracked DMA) ----------------
__device__ __forceinline__ void async_b128_to_lds(void* lds_generic, const void* gsrc) {
    unsigned lds_off = (unsigned)(uintptr_t)lds_generic;  // low 32 bits = LDS offset
    asm volatile("global_load_async_to_lds_b128 %0, %1, off"
                 :: "v"(lds_off), "v"(gsrc) : "memory");
}
// Issue one token row (768 floats = 3 KB): 64 lanes x 16B x 3 channels.
__device__ __forceinline__ void issue_row(float* lds_row, const float* gsrc_row, int tid) {
    async_b128_to_lds(lds_row +           tid * 4, gsrc_row +           tid * 4);
    async_b128_to_lds(lds_row + P2_     + tid * 4, gsrc_row + P2_     + tid * 4);
    async_b128_to_lds(lds_row + 2 * P2_ + tid * 4, gsrc_row + 2 * P2_ + tid * 4);
}

// ---------------- small kernels ----------------
__global__ void k_zero(float* __restrict__ ws, int n) {
    int i = blockIdx.x * blockDim.x + threadIdx.x;
    if (i < n) ws[i] = 0.0f;
}

__global__ void k_counts(const int* __restrict__ ph, const int* __restrict__ pw,
                         const unsigned char* __restrict__ mask,
                         float* __restrict__ counts) {
    int i = blockIdx.x * blockDim.x + threadIdx.x;
    if (i >= N_) return;
    if (mask[i]) return;                       // padded -> weight 0
    int pf = ph[i] * W_ + pw[i];
    atomicAdd(&counts[pf], 1.0f);
}

__global__ void k_nnew(const float* __restrict__ n0, const float* __restrict__ counts,
                       float* __restrict__ n_new) {
    int i = blockIdx.x * blockDim.x + threadIdx.x;
    if (i < HW_) n_new[i] = n0[i] + counts[i];
}

__global__ void k_meannew(const float* __restrict__ mean_old,
                          const float* __restrict__ mean_acc,
                          float* __restrict__ mean_new) {
    int i = blockIdx.x * blockDim.x + threadIdx.x;
    if (i < HW_ * P2_) mean_new[i] = mean_old[i] + mean_acc[i];
}

__global__ void k_invstd(const float* __restrict__ m2_old,
                         const float* __restrict__ m2_acc,
                         const float* __restrict__ n_new,
                         float* __restrict__ invstd) {
    int i = blockIdx.x * blockDim.x + threadIdx.x;
    if (i >= HW_ * P2_) return;
    float nn  = n_new[i >> 8];                 // i / P2_
    float m2n = m2_old[i] + m2_acc[i];
    float var = (nn < 2.0f) ? 1.0f : m2n / fmaxf(nn, 1.0f);
    invstd[i] = 1.0f / (sqrtf(var) + EPSF);
}

// ---------------- pipelined heavy kernels (8 tokens/block, async x-row DMA) ----
// Pattern per iteration u:
//   issue async row u+1 into slot (u+1)&1
//   s_wait_asynccnt <=3  (older row u's 3 asyncs complete; in-order)
//   barrier; consume slot u&1 from LDS; barrier (safe to overwrite next iter)
__global__ __launch_bounds__(64)
void k_accum_mean(const float* __restrict__ x,
                  const int* __restrict__ ph, const int* __restrict__ pw,
                  const unsigned char* __restrict__ mask,
                  const float* __restrict__ mean_old,
                  const float* __restrict__ n_new,
                  float* __restrict__ mean_acc) {
    __shared__ __align__(16) float xs[2][D_];  // 6 KB ping-pong
    int tid  = threadIdx.x;
    int tok0 = blockIdx.x * TPB_TOK;

    issue_row(xs[0], x + (size_t)tok0 * D_, tid);
#pragma unroll 1
    for (int u = 0; u < TPB_TOK; ++u) {
        int tok    = tok0 + u;
        int pf     = ph[tok] * W_ + pw[tok];
        bool valid = (mask[tok] == 0);
        // stat gather straight to registers (reused across channels)
        float4 m  = *reinterpret_cast<const float4*>(mean_old + (size_t)pf * P2_ + tid * 4);
        float  ng = fmaxf(n_new[pf], 1.0f);

        if (u + 1 < TPB_TOK) {
            issue_row(xs[(u + 1) & 1], x + (size_t)(tok + 1) * D_, tid);
            asm volatile("s_wait_asynccnt 0x3" ::: "memory");
        } else {
            asm volatile("s_wait_asynccnt 0x0" ::: "memory");
        }
        __syncthreads();

        const float4* xb = reinterpret_cast<const float4*>(xs[u & 1]);
        float4 x0 = xb[tid], x1 = xb[64 + tid], x2 = xb[128 + tid];
        if (valid) {
            float r   = 1.0f / 3.0f;
            float rng = 1.0f / ng;
            float* dst = mean_acc + (size_t)pf * P2_ + tid * 4;
            atomicAdd(dst + 0, ((x0.x + x1.x + x2.x) * r - m.x) * rng);
            atomicAdd(dst + 1, ((x0.y + x1.y + x2.y) * r - m.y) * rng);
            atomicAdd(dst + 2, ((x0.z + x1.z + x2.z) * r - m.z) * rng);
            atomicAdd(dst + 3, ((x0.w + x1.w + x2.w) * r - m.w) * rng);
        }
        __syncthreads();
    }
}

__global__ __launch_bounds__(64)
void k_accum_m2(const float* __restrict__ x,
                const int* __restrict__ ph, const int* __restrict__ pw,
                const unsigned char* __restrict__ mask,
                const float* __restrict__ mean_old,
                const float* __restrict__ mean_new,
                float* __restrict__ m2_acc) {
    __shared__ __align__(16) float xs[2][D_];
    int tid  = threadIdx.x;
    int tok0 = blockIdx.x * TPB_TOK;

    issue_row(xs[0], x + (size_t)tok0 * D_, tid);
#pragma unroll 1
    for (int u = 0; u < TPB_TOK; ++u) {
        int tok    = tok0 + u;
        int pf     = ph[tok] * W_ + pw[tok];
        bool valid = (mask[tok] == 0);
        float4 mo = *reinterpret_cast<const float4*>(mean_old + (size_t)pf * P2_ + tid * 4);
        float4 mn = *reinterpret_cast<const float4*>(mean_new + (size_t)pf * P2_ + tid * 4);

        if (u + 1 < TPB_TOK) {
            issue_row(xs[(u + 1) & 1], x + (size_t)(tok + 1) * D_, tid);
            asm volatile("s_wait_asynccnt 0x3" ::: "memory");
        } else {
            asm volatile("s_wait_asynccnt 0x0" ::: "memory");
        }
        __syncthreads();

        const float4* xb = reinterpret_cast<const float4*>(xs[u & 1]);
        float4 x0 = xb[tid], x1 = xb[64 + tid], x2 = xb[128 + tid];
        if (valid) {
            float r  = 1.0f / 3.0f;
            float sx = (x0.x + x1.x + x2.x) * r;
            float sy = (x0.y + x1.y + x2.y) * r;
            float sz = (x0.z + x1.z + x2.z) * r;
            float sw = (x0.w + x1.w + x2.w) * r;
            float* dst = m2_acc + (size_t)pf * P2_ + tid * 4;
            atomicAdd(dst + 0, (sx - mo.x) * (sx - mn.x));
            atomicAdd(dst + 1, (sy - mo.y) * (sy - mn.y));
            atomicAdd(dst + 2, (sz - mo.z) * (sz - mn.z));
            atomicAdd(dst + 3, (sw - mo.w) * (sw - mn.w));
        }
        __syncthreads();
    }
}

__global__ __launch_bounds__(64)
void k_normalize(const float* __restrict__ x,
                 const int* __restrict__ ph, const int* __restrict__ pw,
                 const unsigned char* __restrict__ mask,
                 const float* __restrict__ mean_new,
                 const float* __restrict__ invstd,
                 float* __restrict__ out) {
    __shared__ __align__(16) float xs[2][D_];
    int tid  = threadIdx.x;
    int tok0 = blockIdx.x * TPB_TOK;

    issue_row(xs[0], x + (size_t)tok0 * D_, tid);
#pragma unroll 1
    for (int u = 0; u < TPB_TOK; ++u) {
        int tok    = tok0 + u;
        int pf     = ph[tok] * W_ + pw[tok];
        bool valid = (mask[tok] == 0);
        float4 m  = *reinterpret_cast<const float4*>(mean_new + (size_t)pf * P2_ + tid * 4);
        float4 iv = *reinterpret_cast<const float4*>(invstd   + (size_t)pf * P2_ + tid * 4);

        if (u + 1 < TPB_TOK) {
            issue_row(xs[(u + 1) & 1], x + (size_t)(tok + 1) * D_, tid);
            asm volatile("s_wait_asynccnt 0x3" ::: "memory");
        } else {
            asm volatile("s_wait_asynccnt 0x0" ::: "memory");
        }
        __syncthreads();

        float4* orow = reinterpret_cast<float4*>(out + (size_t)tok * D_);
        if (valid) {
            const float4* xb = reinterpret_cast<const float4*>(xs[u & 1]);
#pragma unroll
            for (int c = 0; c < C_; ++c) {
                float4 xv = xb[c * 64 + tid];
                float4 o;
                o.x = (xv.x - m.x) * iv.x;
                o.y = (xv.y - m.y) * iv.y;
                o.z = (xv.z - m.z) * iv.z;
                o.w = (xv.w - m.w) * iv.w;
                orow[c * 64 + tid] = o;
            }
        } else {
            float4 z = {0.f, 0.f, 0.f, 0.f};
            orow[tid] = z; orow[64 + tid] = z; orow[128 + tid] = z;
        }
        __syncthreads();
    }
}

// ---------------- host launcher ----------------
extern "C" void kernel_launch(void* const* d_in, const int* in_sizes, int n_in,
                              void* d_out, int out_size, void* d_ws, size_t ws_size,
                              hipStream_t stream) {
    const float*         patches = (const float*)d_in[0];
    const int*           pos_h   = (const int*)d_in[1];
    const int*           pos_w   = (const int*)d_in[2];
    const unsigned char* padmask = (const unsigned char*)d_in[3];  // jax bool = 1 byte
    const float*         n0      = (const float*)d_in[4];
    const float*         mean0   = (const float*)d_in[5];
    const float*         m2_0    = (const float*)d_in[6];
    float*               out     = (float*)d_out;

    float* ws       = (float*)d_ws;
    float* counts   = ws + WS_COUNTS;
    float* n_new    = ws + WS_NNEW;
    float* mean_acc = ws + WS_MEANACC;
    float* mean_new = ws + WS_MEANNEW;
    float* m2_acc   = ws + WS_M2ACC;
    float* invstd   = ws + WS_INVSTD;

    const int ZT = 256;
    const int HEAVY_GRID = N_ / TPB_TOK;   // 4096 blocks of 64 threads (2 waves)
    k_zero<<<((int)WS_TOTAL + ZT - 1) / ZT, ZT, 0, stream>>>(ws, (int)WS_TOTAL);
    k_counts<<<(N_ + ZT - 1) / ZT, ZT, 0, stream>>>(pos_h, pos_w, padmask, counts);
    k_nnew<<<(HW_ + ZT - 1) / ZT, ZT, 0, stream>>>(n0, counts, n_new);
    k_accum_mean<<<HEAVY_GRID, 64, 0, stream>>>(patches, pos_h, pos_w, padmask,
                                                mean0, n_new, mean_acc);
    k_meannew<<<(HW_ * P2_ + ZT - 1) / ZT, ZT, 0, stream>>>(mean0, mean_acc, mean_new);
    k_accum_m2<<<HEAVY_GRID, 64, 0, stream>>>(patches, pos_h, pos_w, padmask,
                                              mean0, mean_new, m2_acc);
    k_invstd<<<(HW_ * P2_ + ZT - 1) / ZT, ZT, 0, stream>>>(m2_0, m2_acc, n_new, invstd);
    k_normalize<<<HEAVY_GRID, 64, 0, stream>>>(patches, pos_h, pos_w, padmask,
                                               mean_new, invstd, out);
}